// eca_layer_60129542144135
// MI455X (gfx1250) — compile-verified
//
#include <hip/hip_runtime.h>

typedef __attribute__((ext_vector_type(2))) float v2f;
typedef __attribute__((ext_vector_type(4))) float v4f;
typedef __attribute__((ext_vector_type(8))) float v8f;

#define BATCH 32
#define CH    512
#define HW    4096           // 64*64 spatial
#define CTILE 16             // channels per block in stage 1
#define STILE 512            // spatial elements per LDS tile
#define PITCH 528            // LDS row pitch (dwords): decorrelates banks
#define NTILES (HW / STILE)  // 8
#define TOPK  3

typedef __attribute__((address_space(3))) char lds_char;

// ---------------------------------------------------------------------------
// Stage 1: ysum[b][c] = sum over spatial of x[b][c][:]
// Each block: one batch b, 16 consecutive channels. Tiles are streamed into
// LDS with GLOBAL_LOAD_ASYNC_TO_LDS_B128 (CDNA5 async DMA path, ASYNCcnt),
// bypassing VGPR staging entirely. Per-thread accumulation is fixed-order;
// the final 16-partials-per-channel reduction uses V_WMMA_F32_16X16X4_F32
// (B = all-ones => D[m,n] = sum_k A[m,k]) to stay f32-exact.
// ---------------------------------------------------------------------------
__global__ __launch_bounds__(256) void k_spatial_sum(const float* __restrict__ x,
                                                     float* __restrict__ ysum) {
  __shared__ float tile[CTILE * PITCH];
  __shared__ float pex[256];

  const int t   = threadIdx.x;
  const int blk = blockIdx.x;        // 0 .. 1023
  const int b   = blk >> 5;          // 32 channel-tiles per batch
  const int c0  = (blk & 31) * CTILE;
  const float* xb = x + ((size_t)b * CH + c0) * HW;

  const int ch_r = t >> 4;           // read-phase channel (0..15)
  const int jr   = t & 15;           // read-phase sub-slot (0..15)
  float acc = 0.0f;

  for (int ti = 0; ti < NTILES; ++ti) {
    const int s0 = ti * STILE;
    // async cooperative load: 16 channels x 512 floats = 2048 x 16B, straight
    // into LDS (no VGPR staging). GVS mode: SGPR-pair base + per-lane i32
    // byte offset; per-lane LDS byte address in the "vdst" VGPR.
    #pragma unroll
    for (int it = 0; it < 8; ++it) {
      const int l  = t + it * 256;   // float4 index within tile
      const int ch = l >> 7;         // 128 float4 per channel row
      const int sl = l & 127;
      const unsigned goff = (unsigned)((ch * HW + s0 + sl * 4) * 4);
      const unsigned ldsaddr =
          (unsigned)(size_t)(lds_char*)&tile[ch * PITCH + sl * 4];
      asm volatile("global_load_async_to_lds_b128 %0, %1, %2"
                   :
                   : "v"(ldsaddr), "v"(goff), "s"(xb)
                   : "memory");
      if (ti + 1 < NTILES)
        __builtin_prefetch((const char*)xb + goff + STILE * 4, 0, 0);
    }
    asm volatile("s_wait_asynccnt 0x0" ::: "memory");
    __syncthreads();
    // accumulate: thread t sums tile[ch_r][jr + 16k], k = 0..31 (conflict-free)
    const float* row = &tile[ch_r * PITCH];
    #pragma unroll
    for (int k = 0; k < 32; ++k) acc += row[jr + 16 * k];
    __syncthreads();
  }

  pex[t] = acc;                      // 16 partials per channel
  __syncthreads();

  // A-matrix f32 16x4 layout: lanes 0-15 hold {K0,K1} of row m=lane,
  // lanes 16-31 hold {K2,K3} of row m=lane-16. All waves compute the same
  // WMMA so EXEC is all-ones everywhere; only lanes 0/16 of wave 0 store.
  {
    const int m  = t & 15;
    const int hi = (t >> 4) & 1;
    const float* p = &pex[m * 16 + hi * 8];
    v2f a;
    a.x = (p[0] + p[1]) + (p[2] + p[3]);
    a.y = (p[4] + p[5]) + (p[6] + p[7]);
    v2f bones; bones.x = 1.0f; bones.y = 1.0f;
    v8f cz = {};
    v8f dm = __builtin_amdgcn_wmma_f32_16x16x4_f32(
        /*neg_a=*/false, a, /*neg_b=*/false, bones,
        /*c_mod=*/(short)0, cz, /*reuse_a=*/false, /*reuse_b=*/false);
    // D layout: lane 0 holds D[0..7][0] in dm[0..7]; lane 16 holds D[8..15][0].
    if (t < 32 && m == 0) {
      float* outp = &ysum[b * CH + c0 + hi * 8];
      #pragma unroll
      for (int r = 0; r < 8; ++r) outp[r] = dm[r];
    }
  }
}

// ---------------------------------------------------------------------------
// Stage 2: 3-tap conv over channels (zero pad) on y = ysum/4096; sigmoid is
// monotonic so top-3 of the conv output == top-3 after sigmoid. Serial
// argmax x3 with lower-index tie-break (matches lax.top_k, deterministic).
// ---------------------------------------------------------------------------
__global__ __launch_bounds__(CH) void k_conv_topk(const float* __restrict__ ysum,
                                                  const float* __restrict__ w,
                                                  int* __restrict__ idx) {
  __shared__ float tch[CH];
  const int b = blockIdx.x;
  const int c = threadIdx.x;
  const float inv = 1.0f / (float)HW;
  const float ym1 = (c > 0)      ? ysum[b * CH + c - 1] * inv : 0.0f;
  const float y0  =                ysum[b * CH + c]     * inv;
  const float yp1 = (c < CH - 1) ? ysum[b * CH + c + 1] * inv : 0.0f;
  tch[c] = w[0] * ym1 + w[1] * y0 + w[2] * yp1;
  __syncthreads();
  if (c == 0) {
    int taken0 = -1, taken1 = -1;
    for (int j = 0; j < TOPK; ++j) {
      float best = -__builtin_inff();
      int bi = 0;
      for (int i = 0; i < CH; ++i) {
        if (i == taken0 || i == taken1) continue;
        const float v = tch[i];
        if (v > best) { best = v; bi = i; }
      }
      idx[b * TOPK + j] = bi;
      if (j == 0) taken0 = bi; else if (j == 1) taken1 = bi;
    }
  }
}

// ---------------------------------------------------------------------------
// Stage 3: out[b][j][:] = x[b][idx[b][j]][:]  (mask multiply is identity at
// the gathered indices). 96 blocks x 256 threads, float4 copies.
// ---------------------------------------------------------------------------
__global__ __launch_bounds__(256) void k_gather(const float* __restrict__ x,
                                                const int* __restrict__ idx,
                                                float* __restrict__ out) {
  const int blk = blockIdx.x;          // b*TOPK + j
  const int b = blk / TOPK;
  const int j = blk - b * TOPK;
  const int c = idx[b * TOPK + j];
  const float* src = x + ((size_t)b * CH + c) * HW;
  float* dst = out + (size_t)blk * HW;
  const int t = threadIdx.x;
  #pragma unroll
  for (int i = 0; i < 4; ++i) {
    const int o = (t + i * 256) * 4;
    *(v4f*)(dst + o) = *(const v4f*)(src + o);
  }
}

extern "C" void kernel_launch(void* const* d_in, const int* in_sizes, int n_in,
                              void* d_out, int out_size, void* d_ws, size_t ws_size,
                              hipStream_t stream) {
  const float* x = (const float*)d_in[0];     // (32, 512, 64, 64) f32
  const float* w = (const float*)d_in[1];     // (1, 1, 3) f32
  float* out  = (float*)d_out;                // (32, 3, 64, 64) f32
  float* ysum = (float*)d_ws;                                   // 32*512 f32
  int*   idx  = (int*)((char*)d_ws + BATCH * CH * sizeof(float)); // 32*3 int

  k_spatial_sum<<<BATCH * (CH / CTILE), 256, 0, stream>>>(x, ysum);
  k_conv_topk<<<BATCH, CH, 0, stream>>>(ysum, w, idx);
  k_gather<<<BATCH * TOPK, 256, 0, stream>>>(x, idx, out);
}